// WKV_13099650253092
// MI455X (gfx1250) — compile-verified
//
#include <hip/hip_runtime.h>

// Problem constants (B,T,D) = (4,4096,512)
#define B_   4
#define T_   4096
#define D_   512
#define TM1  4095          // T-1: length of the weighted-history sum
#define NC   32            // scan chunks
#define CHUNK 128          // TM1 = 31*128 + 127
#define KT   32            // K-slab depth for GEMMs
#define LDB  68            // padded LDS stride for [KT][64] slabs
#define LDA2 36            // padded LDS stride for [64][KT] A slab (gemm2)
#define NSLAB1 ((TM1 + KT - 1) / KT)   // 128
#define NSLAB2 (D_ / KT)               // 16

typedef __attribute__((ext_vector_type(2))) float v2f;
typedef __attribute__((ext_vector_type(4))) float v4f;
typedef __attribute__((ext_vector_type(8))) float v8f;
typedef __attribute__((ext_vector_type(4))) int   v4i;

// ---- CDNA5 async global->LDS (ASYNCcnt) with synchronous fallback ---------
#if defined(__has_builtin)
#if __has_builtin(__builtin_amdgcn_global_load_async_to_lds_b128) && \
    __has_builtin(__builtin_amdgcn_s_wait_asynccnt)
#define USE_ASYNC_LDS 1
#endif
#endif
#ifndef USE_ASYNC_LDS
#define USE_ASYNC_LDS 0
#endif

__device__ __forceinline__ void async_ld16(void* lds, const void* g) {
#if USE_ASYNC_LDS
  __builtin_amdgcn_global_load_async_to_lds_b128(
      (__attribute__((address_space(1))) v4i*)(void*)g,
      (__attribute__((address_space(3))) v4i*)lds, 0, 0);
#else
  *(v4f*)lds = *(const v4f*)g;
#endif
}

template <int N>
__device__ __forceinline__ void wait_async() {
#if USE_ASYNC_LDS
  __builtin_amdgcn_s_wait_asynccnt(N);
#endif
}

// D = A(16x4, f32) * B(4x16, f32) + C(16x16, f32)   -> v_wmma_f32_16x16x4_f32
__device__ __forceinline__ v8f wmma4(v2f a, v2f b, v8f c) {
  return __builtin_amdgcn_wmma_f32_16x16x4_f32(false, a, false, b, (short)0, c,
                                               false, false);
}

// ---------------- Pass A: per-chunk products of w over time -----------------
__global__ void chunk_prod_kernel(const float* __restrict__ w,
                                  float* __restrict__ cprod) {
  const int d = blockIdx.x * 256 + threadIdx.x;
  const int c = blockIdx.y;
  const int b = blockIdx.z;
  const int lo = c * CHUNK;
  const int hi = (lo + CHUNK < TM1) ? (lo + CHUNK) : TM1;
  const float* wp = w + ((size_t)b * T_ + lo) * D_ + d;
  float p = 1.0f;
  for (int s = lo; s < hi; ++s) { p *= *wp; wp += D_; }
  cprod[((size_t)b * NC + c) * D_ + d] = p;
}

// ------------- Pass B: exclusive suffix scan over chunk products ------------
__global__ void chunk_suffix_kernel(const float* __restrict__ cprod,
                                    float* __restrict__ S) {
  const int d = blockIdx.x * 256 + threadIdx.x;
  const int b = blockIdx.y;
  float run = 1.0f;
  for (int c = NC - 1; c >= 0; --c) {
    const size_t i = ((size_t)b * NC + c) * D_ + d;
    S[i] = run;
    run *= cprod[i];
  }
}

// ------ Pass C: cpv[b,t,d] = (exclusive suffix prod of w) * v[b,t,d] --------
__global__ void cpv_kernel(const float* __restrict__ w,
                           const float* __restrict__ v,
                           const float* __restrict__ S,
                           float* __restrict__ cpv) {
  const int d = blockIdx.x * 256 + threadIdx.x;
  const int c = blockIdx.y;
  const int b = blockIdx.z;
  const int lo = c * CHUNK;
  const int hi = (lo + CHUNK < TM1) ? (lo + CHUNK) : TM1;
  float run = S[((size_t)b * NC + c) * D_ + d];   // prod over all later chunks
  for (int s = hi - 1; s >= lo; --s) {
    const size_t gi = ((size_t)b * T_ + s) * D_ + d;
    cpv[((size_t)b * TM1 + s) * D_ + d] = run * v[gi];
    run *= w[gi];   // now run = prod_{s' >= s}, i.e. cp for t = s-1
  }
}

// ------------- 64-float zero pad used for the GEMM1 K tail ------------------
__global__ void zeropad_kernel(float* __restrict__ zp) {
  zp[threadIdx.x] = 0.0f;
}

// --------- GEMM1: wkv[b,j,d] = sum_t k[b,t,j]*cpv[b,t,d] + rank-1 -----------
__global__ void __launch_bounds__(256)
gemm1_kernel(const float* __restrict__ k, const float* __restrict__ cpv,
             const float* __restrict__ v, const float* __restrict__ u,
             const float* __restrict__ zp, float* __restrict__ wkv) {
  __shared__ float As[2][KT][LDB];   // As[p][kk][j] = k[b, t0+kk, j0+j]
  __shared__ float Bs[2][KT][LDB];   // Bs[p][kk][d] = cpv[b, t0+kk, d0+d]
  const int tid  = threadIdx.x;
  const int b    = blockIdx.z;
  const int j0   = blockIdx.x * 64;
  const int d0   = blockIdx.y * 64;
  const int wave = tid >> 5;
  const int lane = tid & 31;
  const int half = lane >> 4;     // lane half selects K pair per ISA layout
  const int l16  = lane & 15;
  const int sj   = (wave & 3) * 16;          // this wave's j subtile
  const int sd0  = (wave >> 2) * 32;         // this wave's first d subtile
  const int lj   = (tid & 15) * 4;           // slab-load column (float4/b128)
  const int lk   = tid >> 4;                 // slab-load row

  v8f acc0 = {0.f,0.f,0.f,0.f,0.f,0.f,0.f,0.f};
  v8f acc1 = {0.f,0.f,0.f,0.f,0.f,0.f,0.f,0.f};

  // Issue one K-slab's loads (4 async b128 ops per thread) into buffer p.
  auto issue = [&](int slab, int p) {
    const int t0 = slab * KT;
#pragma unroll
    for (int kh = 0; kh < 2; ++kh) {
      const int kk = lk + kh * 16;
      const int t  = t0 + kk;
      const float* ap = zp + lj;             // zero row for the K tail
      const float* bp = zp + lj;
      if (t < TM1) {
        ap = k   + ((size_t)b * T_  + t) * D_ + j0 + lj;
        bp = cpv + ((size_t)b * TM1 + t) * D_ + d0 + lj;
      }
      async_ld16(&As[p][kk][lj], ap);
      async_ld16(&Bs[p][kk][lj], bp);
    }
  };

  issue(0, 0);                               // prologue
  for (int s = 0; s < NSLAB1; ++s) {
    const int cur = s & 1;
    const bool more = (s + 1) < NSLAB1;
    if (more) {
      issue(s + 1, cur ^ 1);                 // prefetch next slab (other buf)
      wait_async<4>();                       // retire slab s's 4 ops
    } else {
      wait_async<0>();
    }
    __syncthreads();
#pragma unroll
    for (int kk = 0; kk < KT; kk += 4) {
      v2f a, b0v, b1v;
      a.x   = As[cur][kk + 2*half    ][sj + l16];
      a.y   = As[cur][kk + 2*half + 1][sj + l16];
      b0v.x = Bs[cur][kk + 2*half    ][sd0 + l16];
      b0v.y = Bs[cur][kk + 2*half + 1][sd0 + l16];
      b1v.x = Bs[cur][kk + 2*half    ][sd0 + 16 + l16];
      b1v.y = Bs[cur][kk + 2*half + 1][sd0 + 16 + l16];
      acc0 = wmma4(a, b0v, acc0);
      acc1 = wmma4(a, b1v, acc1);
    }
    __syncthreads();                         // buf reusable for slab s+2
  }

  // Epilogue: += u[j] * k[b,T-1,j] * v[b,T-1,d]; store wkv
  const size_t lastRow = ((size_t)b * T_ + TM1) * D_;
  const float vlast0 = v[lastRow + d0 + sd0 + l16];
  const float vlast1 = v[lastRow + d0 + sd0 + 16 + l16];
  float* wkvb = wkv + (size_t)b * D_ * D_;
#pragma unroll
  for (int i = 0; i < 8; ++i) {              // C/D layout: VGPR i -> rows i, i+8
    const int j = j0 + sj + i + half * 8;
    const float uk = u[j] * k[lastRow + j];
    wkvb[(size_t)j * D_ + d0 + sd0      + l16] = acc0[i] + uk * vlast0;
    wkvb[(size_t)j * D_ + d0 + sd0 + 16 + l16] = acc1[i] + uk * vlast1;
  }
}

// ------------- GEMM2: out[b,t,d] = sum_j r[b,t,j] * wkv[b,j,d] --------------
__global__ void __launch_bounds__(256)
gemm2_kernel(const float* __restrict__ r, const float* __restrict__ wkv,
             float* __restrict__ out) {
  __shared__ float As[2][64][LDA2];  // As[p][m][kk] = r[b, t0+m, jj0+kk]
  __shared__ float Bs[2][KT][LDB];   // Bs[p][kk][d] = wkv[b, jj0+kk, d0+d]
  const int tid  = threadIdx.x;
  const int b    = blockIdx.z;
  const int t0   = blockIdx.x * 64;
  const int d0   = blockIdx.y * 64;
  const int wave = tid >> 5;
  const int lane = tid & 31;
  const int half = lane >> 4;
  const int l16  = lane & 15;
  const int st   = (wave & 3) * 16;          // t subtile
  const int sd0  = (wave >> 2) * 32;         // first d subtile
  const int am   = tid >> 3;                 // A-load row  (0..31, +32)
  const int ak   = (tid & 7) * 4;            // A-load col  (b128 along j)
  const int lj   = (tid & 15) * 4;           // B-load col
  const int lk   = tid >> 4;                 // B-load row

  v8f acc0 = {0.f,0.f,0.f,0.f,0.f,0.f,0.f,0.f};
  v8f acc1 = {0.f,0.f,0.f,0.f,0.f,0.f,0.f,0.f};

  auto issue = [&](int slab, int p) {
    const int jj0 = slab * KT;
#pragma unroll
    for (int mh = 0; mh < 2; ++mh) {
      const int m = am + mh * 32;
      async_ld16(&As[p][m][ak],
                 r + ((size_t)b * T_ + t0 + m) * D_ + jj0 + ak);
    }
#pragma unroll
    for (int kh = 0; kh < 2; ++kh) {
      const int kk = lk + kh * 16;
      async_ld16(&Bs[p][kk][lj],
                 wkv + ((size_t)b * D_ + jj0 + kk) * D_ + d0 + lj);
    }
  };

  issue(0, 0);
  for (int s = 0; s < NSLAB2; ++s) {
    const int cur = s & 1;
    const bool more = (s + 1) < NSLAB2;
    if (more) {
      issue(s + 1, cur ^ 1);
      wait_async<4>();
    } else {
      wait_async<0>();
    }
    __syncthreads();
#pragma unroll
    for (int kk = 0; kk < KT; kk += 4) {
      v2f a, b0v, b1v;
      a.x   = As[cur][st + l16][kk + 2*half    ];
      a.y   = As[cur][st + l16][kk + 2*half + 1];
      b0v.x = Bs[cur][kk + 2*half    ][sd0 + l16];
      b0v.y = Bs[cur][kk + 2*half + 1][sd0 + l16];
      b1v.x = Bs[cur][kk + 2*half    ][sd0 + 16 + l16];
      b1v.y = Bs[cur][kk + 2*half + 1][sd0 + 16 + l16];
      acc0 = wmma4(a, b0v, acc0);
      acc1 = wmma4(a, b1v, acc1);
    }
    __syncthreads();
  }

#pragma unroll
  for (int i = 0; i < 8; ++i) {
    const int t = t0 + st + i + half * 8;
    out[((size_t)b * T_ + t) * D_ + d0 + sd0      + l16] = acc0[i];
    out[((size_t)b * T_ + t) * D_ + d0 + sd0 + 16 + l16] = acc1[i];
  }
}

extern "C" void kernel_launch(void* const* d_in, const int* in_sizes, int n_in,
                              void* d_out, int out_size, void* d_ws, size_t ws_size,
                              hipStream_t stream) {
  const float* r = (const float*)d_in[0];
  const float* w = (const float*)d_in[1];
  const float* k = (const float*)d_in[2];
  const float* v = (const float*)d_in[3];
  const float* u = (const float*)d_in[4];
  float* out = (float*)d_out;

  // Workspace (floats): cpv | chunkProd | chunkSuffix | wkv | zeropad (~38.3MB)
  float* ws    = (float*)d_ws;
  float* cpv   = ws;                                   // B*TM1*D
  float* cprod = cpv   + (size_t)B_ * TM1 * D_;        // B*NC*D
  float* S     = cprod + (size_t)B_ * NC  * D_;        // B*NC*D
  float* wkv   = S     + (size_t)B_ * NC  * D_;        // B*D*D
  float* zp    = wkv   + (size_t)B_ * D_  * D_;        // 64 zeros

  zeropad_kernel     <<<1, 64, 0, stream>>>(zp);
  chunk_prod_kernel  <<<dim3(D_/256, NC, B_), 256, 0, stream>>>(w, cprod);
  chunk_suffix_kernel<<<dim3(D_/256, B_),     256, 0, stream>>>(cprod, S);
  cpv_kernel         <<<dim3(D_/256, NC, B_), 256, 0, stream>>>(w, v, S, cpv);
  gemm1_kernel       <<<dim3(D_/64, D_/64, B_), 256, 0, stream>>>(k, cpv, v, u, zp, wkv);
  gemm2_kernel       <<<dim3(T_/64, D_/64, B_), 256, 0, stream>>>(r, wkv, out);
}